// Region_52527450030759
// MI455X (gfx1250) — compile-verified
//
#include <hip/hip_runtime.h>
#include <math.h>

// Problem constants (match reference)
#define BATCH 1024
#define KC    7
#define NNODE 32
#define FF    5
#define EE    16
#define COF   50
#define COK   128
#define COE   32
#define XRD   50   // per-cluster output feature dim
#define ODIM  60   // coarse output feature dim

#define USE_ASYNC_LDS 1

typedef float v2f_t __attribute__((ext_vector_type(2)));
typedef float v8f_t __attribute__((ext_vector_type(8)));

// ---------------- scalar helpers ----------------
__device__ __forceinline__ float eluf(float x){ return x > 0.f ? x : expm1f(x); }
__device__ __forceinline__ float seluf(float x){
  const float a = 1.6732632423543772f, s = 1.0507009873554805f;
  return x > 0.f ? s*x : s*a*expm1f(x);
}
__device__ __forceinline__ float sigm(float x){ return 1.f/(1.f+expf(-x)); }

__device__ __forceinline__ float wave_sum(float v){
  #pragma unroll
  for (int o=16;o>0;o>>=1) v += __shfl_xor(v,o,32);
  return v;
}
__device__ __forceinline__ float wave_max(float v){
  #pragma unroll
  for (int o=16;o>0;o>>=1) v = fmaxf(v, __shfl_xor(v,o,32));
  return v;
}

// ---------------- WMMA f32 16x16x4 helpers (wave32) ----------------
// A 16x4: lanes 0-15 hold M=0..15; VGPR0=K(2*half), VGPR1=K(2*half+1)
__device__ __forceinline__ v2f_t lda_frag(const float* M, int ld, int mbase, int kbase, int lane){
  int half = lane>>4, r = lane&15;
  v2f_t a; a.x = M[(mbase+r)*ld + kbase + 2*half];
           a.y = M[(mbase+r)*ld + kbase + 2*half + 1];
  return a;
}
// B 4x16: row K striped across lanes; VGPR0=K(2*half), VGPR1=K(2*half+1), col = lane&15
__device__ __forceinline__ v2f_t ldb_frag(const float* M, int ld, int kbase, int nbase, int lane){
  int half = lane>>4, c = lane&15;
  v2f_t b; b.x = M[(kbase + 2*half    )*ld + nbase + c];
           b.y = M[(kbase + 2*half + 1)*ld + nbase + c];
  return b;
}
__device__ __forceinline__ v8f_t wmma4(v2f_t a, v2f_t b, v8f_t c){
  return __builtin_amdgcn_wmma_f32_16x16x4_f32(false, a, false, b, (short)0, c, false, false);
}
__device__ __forceinline__ v8f_t zero8(){
  v8f_t z = {0.f,0.f,0.f,0.f,0.f,0.f,0.f,0.f};
  return z;
}

#if USE_ASYNC_LDS
// gfx1250 async global->LDS copy (ASYNCcnt path). LDS byte address obtained via
// addrspace(3) cast (AS3 pointers are 32-bit LDS offsets on AMDGPU).
typedef __attribute__((address_space(3))) float lds_float_t;
__device__ __forceinline__ unsigned lds_addr_of(const void* p){
  return (unsigned)(unsigned long long)(lds_float_t*)(p);
}
__device__ __forceinline__ void async_copy_b32(void* lds_dst, const float* gsrc){
  unsigned dst = lds_addr_of(lds_dst);
  asm volatile("global_load_async_to_lds_b32 %0, %1, off"
               :: "v"(dst), "v"(gsrc) : "memory");
}
__device__ __forceinline__ void async_wait0(){
  asm volatile("s_wait_asynccnt 0" ::: "memory");
}
#endif

// ---------------- CBAM (C=32 channels), data in LDS, one wave ----------------
// scr: avg[32] mx[32] h[8] sm[52] sx[52] sg[52]  (>= 228 floats)
__device__ void cbam_c32(float* X, int ld, int D,
                         const float* w1, const float* w2, const float* cv,
                         float* scr, int lane){
  float* s_avg = scr;      float* s_mx = scr+32;  float* s_h  = scr+64;
  float* s_sm  = scr+72;   float* s_sx = scr+124; float* s_sg = scr+176;
  float av=0.f, mx=-3.4e38f;
  for (int d=0; d<D; ++d){ float v = X[lane*ld+d]; av+=v; mx=fmaxf(mx,v); }
  s_avg[lane]=av/(float)D; s_mx[lane]=mx;
  __syncthreads();
  if (lane<8){
    float hA=0.f,hM=0.f;
    for (int c=0;c<32;++c){ hA += s_avg[c]*w1[c*8+lane]; hM += s_mx[c]*w1[c*8+lane]; }
    s_h[lane] = fmaxf(hA,0.f)+fmaxf(hM,0.f);
  }
  __syncthreads();
  float ca=0.f;
  for (int j=0;j<8;++j) ca += s_h[j]*w2[j*32+lane];
  ca = sigm(ca);
  for (int d=0; d<D; ++d) X[lane*ld+d]*=ca;
  __syncthreads();
  for (int d=lane; d<D; d+=32){
    float sm=0.f, sx=-3.4e38f;
    for (int c=0;c<32;++c){ float v=X[c*ld+d]; sm+=v; sx=fmaxf(sx,v); }
    s_sm[d]=sm*(1.f/32.f); s_sx[d]=sx;
  }
  __syncthreads();
  for (int d=lane; d<D; d+=32){
    float acc=0.f;
    for (int t=0;t<7;++t){ int p=d+t-3; if (p>=0&&p<D) acc += cv[t]*s_sm[p]+cv[7+t]*s_sx[p]; }
    s_sg[d]=sigm(acc);
  }
  __syncthreads();
  for (int d=0; d<D; ++d) X[lane*ld+d]*=s_sg[d];
  __syncthreads();
}

// ---------------- CBAM (C=7 channels) ----------------
// scr: avg[8] mx[8] h[4] sm[64] sx[64] sg[64] (>= 212 floats)
__device__ void cbam_c7(float* X, int ld, int D,
                        const float* w1, const float* w2, const float* cv,
                        float* scr, int lane){
  float* s_avg=scr; float* s_mx=scr+8; float* s_h=scr+16;
  float* s_sm=scr+20; float* s_sx=scr+84; float* s_sg=scr+148;
  if (lane<7){
    float av=0.f,mx=-3.4e38f;
    for (int d=0;d<D;++d){ float v=X[lane*ld+d]; av+=v; mx=fmaxf(mx,v); }
    s_avg[lane]=av/(float)D; s_mx[lane]=mx;
  }
  __syncthreads();
  if (lane==0){
    float hA=0.f,hM=0.f;
    for (int c=0;c<7;++c){ hA+=s_avg[c]*w1[c]; hM+=s_mx[c]*w1[c]; }
    s_h[0]=fmaxf(hA,0.f)+fmaxf(hM,0.f);
  }
  __syncthreads();
  if (lane<7){
    float ca=sigm(s_h[0]*w2[lane]);
    for (int d=0;d<D;++d) X[lane*ld+d]*=ca;
  }
  __syncthreads();
  for (int d=lane; d<D; d+=32){
    float sm=0.f,sx=-3.4e38f;
    for (int c=0;c<7;++c){ float v=X[c*ld+d]; sm+=v; sx=fmaxf(sx,v); }
    s_sm[d]=sm*(1.f/7.f); s_sx[d]=sx;
  }
  __syncthreads();
  for (int d=lane; d<D; d+=32){
    float acc=0.f;
    for (int t=0;t<7;++t){ int p=d+t-3; if(p>=0&&p<D) acc+=cv[t]*s_sm[p]+cv[7+t]*s_sx[p]; }
    s_sg[d]=sigm(acc);
  }
  __syncthreads();
  if (lane<7){ for (int d=0;d<D;++d) X[lane*ld+d]*=s_sg[d]; }
  __syncthreads();
}

// ================= K1: per-(cluster,batch) fine pipeline =================
// one wave32 per (b,k): get_adj_f (adj + VQ + SE) then res_gcn_f -> xr[b][k][32][50]
// All M=32 GEMMs (p@sub, t@theta, ze=adj@win, -2ze@cbT, recon=Q@wout, adj@xw) use
// V_WMMA_F32_16X16X4_F32 on LDS-staged operands.
__global__ __launch_bounds__(32)
void k1_cluster(const float* __restrict__ x,
                const float* __restrict__ ga_p,   const float* __restrict__ ga_bias,
                const float* __restrict__ ga_q,   const float* __restrict__ ga_theta,
                const float* __restrict__ ga_se_w1,const float* __restrict__ ga_se_w2,
                const float* __restrict__ ga_cb,  const float* __restrict__ ga_win,
                const float* __restrict__ ga_wout,
                const float* __restrict__ rg_w1,  const float* __restrict__ rg_w2,
                const float* __restrict__ cb1_w1, const float* __restrict__ cb1_w2, const float* __restrict__ cb1_cv,
                const float* __restrict__ cb2_w1, const float* __restrict__ cb2_w2, const float* __restrict__ cb2_cv,
                float* __restrict__ xr_out, float* __restrict__ l1_out)
{
  const int lane = threadIdx.x;
  const int b = blockIdx.x, k = blockIdx.y;

  __shared__ float s_sub[NNODE*16];          // node features padded [32][16]
  __shared__ float s_thetaP[8*NNODE*FF];     // theta padded [8][160]
  __shared__ float s_adj[FF*NNODE*NNODE];    // adj channels, later recon
  __shared__ float s_win[NNODE*EE];
  __shared__ float s_cb[NNODE*EE];
  __shared__ float s_cbT[EE*NNODE];          // codebook^T [16][32]
  __shared__ float s_wout[EE*NNODE];
  __shared__ float s_cbsq[NNODE];
  __shared__ float s_ze[NNODE*EE];           // o-staging, later ze per channel
  __shared__ float s_zez[NNODE];
  __shared__ int   s_bidx[NNODE];
  __shared__ float s_q[NNODE*EE];            // gathered code rows
  __shared__ float s_tpad[NNODE*8];          // t padded [32][8]
  __shared__ float s_A[NNODE*NNODE];         // p staging, then final adjacency
  __shared__ float s_xw[NNODE*NNODE];        // padded GEMM B operand
  __shared__ float s_cat[NNODE*52];          // concat feature buffer
  __shared__ float s_scr[232];

  // stage per-cluster constants
  const float* theta_k = ga_theta + k*FF*NNODE*FF;
  __builtin_prefetch(theta_k, 0, 1);
  __builtin_prefetch(ga_cb + k*NNODE*EE, 0, 1);
  for (int i=lane;i<8*NNODE*FF;i+=32){
    int r=i/(NNODE*FF), c=i%(NNODE*FF);
    s_thetaP[i] = (r<FF) ? theta_k[r*(NNODE*FF)+c] : 0.f;
  }
  for (int i=lane;i<NNODE*EE;i+=32){
    s_win[i]  = ga_win [k*NNODE*EE+i];
    s_cb[i]   = ga_cb  [k*NNODE*EE+i];
    s_wout[i] = ga_wout[k*EE*NNODE+i];
  }
  for (int i=lane;i<EE*NNODE;i+=32){
    int e=i/NNODE, j=i%NNODE;
    s_cbT[i] = ga_cb[k*NNODE*EE + j*EE + e];
  }
  {
    const float* cr = ga_cb + k*NNODE*EE + lane*EE;
    float a=0.f;
    for (int e=0;e<EE;++e) a+=cr[e]*cr[e];
    s_cbsq[lane]=a;
  }
  for (int i=lane;i<NNODE*NNODE;i+=32) s_A[i]=ga_p[k*NNODE*NNODE+i]; // temp: p
  // node row for this lane (padded to 16 cols)
  float xr5[FF];
  {
    const float* xp = x + ((size_t)b*(KC*NNODE) + k*NNODE + lane)*FF;
    for (int f=0;f<FF;++f){ xr5[f]=xp[f]; s_sub[lane*16+f]=xr5[f]; }
    for (int f=FF;f<16;++f) s_sub[lane*16+f]=0.f;
  }
  __syncthreads();

  // ---- o = p@sub + bias via WMMA (32x32x16), result to s_ze[32][16] ----
  for (int mt=0;mt<2;++mt){
    v8f_t acc = zero8();
    for (int kk=0;kk<NNODE;kk+=4){
      v2f_t a = lda_frag(s_A,   NNODE, mt*16, kk, lane);
      v2f_t bb= ldb_frag(s_sub, 16,    kk,    0,  lane);
      acc = wmma4(a,bb,acc);
    }
    int half=lane>>4, col=lane&15;
    #pragma unroll
    for (int i=0;i<8;++i){
      int row=mt*16+i+8*half;
      float v=acc[i];
      if (col<FF) v += ga_bias[(k*NNODE+row)*FF+col];
      s_ze[row*EE+col]=v;
    }
  }
  __syncthreads();

  // t = o@q (row per lane), pad to 8 for WMMA
  float t5[FF];
  {
    float o5[FF];
    for (int f=0;f<FF;++f) o5[f]=s_ze[lane*EE+f];
    const float* q = ga_q + k*FF*FF;
    for (int c=0;c<FF;++c){ float a=0.f; for (int f=0;f<FF;++f) a+=o5[f]*q[f*FF+c]; t5[c]=a; }
    for (int c=0;c<8;++c) s_tpad[lane*8+c] = (c<FF)? t5[c] : 0.f;
  }
  __syncthreads();

  // ---- g = elu(t@theta) via WMMA (32x8x160) scattered to adj[f][n][m] ----
  for (int nt=0;nt<10;++nt){
    for (int mt=0;mt<2;++mt){
      v8f_t acc = zero8();
      for (int kk=0;kk<8;kk+=4){
        v2f_t a = lda_frag(s_tpad,   8,           mt*16, kk,    lane);
        v2f_t bb= ldb_frag(s_thetaP, NNODE*FF,    kk,    nt*16, lane);
        acc = wmma4(a,bb,acc);
      }
      int half=lane>>4, col=lane&15;
      #pragma unroll
      for (int i=0;i<8;++i){
        int n=mt*16+i+8*half, j=nt*16+col;
        int m=j/FF, ff=j%FF;
        s_adj[ff*NNODE*NNODE + n*NNODE + m] = eluf(acc[i]);
      }
    }
  }
  __syncthreads();
  // L1 normalize each (f, n=lane) row over m
  {
    float inv[FF];
    for (int f=0;f<FF;++f){
      const float* row=&s_adj[f*NNODE*NNODE+lane*NNODE];
      float nr=0.f;
      for (int m=0;m<NNODE;++m) nr+=fabsf(row[m]);
      inv[f]=1.f/(nr+1e-6f);
    }
    for (int f=0;f<FF;++f){
      float* row=&s_adj[f*NNODE*NNODE+lane*NNODE];
      for (int m=0;m<NNODE;++m) row[m]*=inv[f];
    }
  }
  __syncthreads();

  // ---- VQ per channel, fully WMMA: ze -> argmin distance -> recon ----
  float lossacc=0.f, fmean[FF];
  for (int f=0;f<FF;++f){
    // ze = adj[f] @ win (32x32x16)
    for (int mt=0;mt<2;++mt){
      v8f_t acc = zero8();
      for (int kk=0;kk<NNODE;kk+=4){
        v2f_t a = lda_frag(s_adj + f*NNODE*NNODE, NNODE, mt*16, kk, lane);
        v2f_t bb= ldb_frag(s_win, EE, kk, 0, lane);
        acc = wmma4(a,bb,acc);
      }
      int half=lane>>4, col=lane&15;
      #pragma unroll
      for (int i=0;i<8;++i) s_ze[(mt*16+i+8*half)*EE+col]=acc[i];
    }
    __syncthreads();
    {
      const float* zr=&s_ze[lane*EE];
      float z=0.f;
      for (int e=0;e<EE;++e) z+=zr[e]*zr[e];
      s_zez[lane]=z;
    }
    __syncthreads();
    // d2 = ||ze||^2 - 2*(ze@cbT) + ||cb||^2 ; argmin over 32 codes (32x16x32)
    for (int mt=0;mt<2;++mt){
      float bd[8]; int bj[8];
      #pragma unroll
      for (int i=0;i<8;++i){ bd[i]=3.4e38f; bj[i]=0; }
      int half=lane>>4, col=lane&15;
      for (int nt=0;nt<2;++nt){
        v8f_t acc = zero8();
        for (int kk=0;kk<EE;kk+=4){
          v2f_t a = lda_frag(s_ze,  EE,    mt*16, kk,    lane);
          v2f_t bb= ldb_frag(s_cbT, NNODE, kk,    nt*16, lane);
          acc = wmma4(a,bb,acc);
        }
        int n=nt*16+col;
        #pragma unroll
        for (int i=0;i<8;++i){
          float d2 = s_zez[mt*16+i+8*half] - 2.f*acc[i] + s_cbsq[n];
          if (d2<bd[i] || (d2==bd[i] && n<bj[i])){ bd[i]=d2; bj[i]=n; }
        }
      }
      #pragma unroll
      for (int off=1; off<16; off<<=1){
        #pragma unroll
        for (int i=0;i<8;++i){
          float od=__shfl_xor(bd[i],off,32);
          int   oj=__shfl_xor(bj[i],off,32);
          if (od<bd[i] || (od==bd[i]&&oj<bj[i])){ bd[i]=od; bj[i]=oj; }
        }
      }
      if (col==0){
        #pragma unroll
        for (int i=0;i<8;++i){
          s_bidx[mt*16+i+8*half]=bj[i];
          lossacc += bd[i];                 // SSE of that row
        }
      }
    }
    __syncthreads();
    // gather selected code rows: Q[n] = cb[idx[n]]
    {
      int j=s_bidx[lane];
      for (int e=0;e<EE;++e) s_q[lane*EE+e]=s_cb[j*EE+e];
    }
    __syncthreads();
    // recon = Q @ wout (32x16x32), overwrite adj[f]; accumulate channel mean
    float fsum=0.f;
    for (int nt=0;nt<2;++nt){
      for (int mt=0;mt<2;++mt){
        v8f_t acc = zero8();
        for (int kk=0;kk<EE;kk+=4){
          v2f_t a = lda_frag(s_q,    EE,    mt*16, kk,    lane);
          v2f_t bb= ldb_frag(s_wout, NNODE, kk,    nt*16, lane);
          acc = wmma4(a,bb,acc);
        }
        int half=lane>>4, col=lane&15;
        #pragma unroll
        for (int i=0;i<8;++i){
          float v=acc[i];
          s_adj[f*NNODE*NNODE + (mt*16+i+8*half)*NNODE + nt*16+col]=v;
          fsum+=v;
        }
      }
    }
    fmean[f]=wave_sum(fsum)*(1.f/(NNODE*NNODE));
    __syncthreads();
  }
  {
    float tot = wave_sum(lossacc);
    if (lane==0) l1_out[k*BATCH+b]=tot;
  }

  // SE gate (uniform over wave)
  float g5[FF];
  {
    const float* sw1 = ga_se_w1 + k*FF*2;
    const float* sw2 = ga_se_w2 + k*2*FF;
    float h0=0.f,h1=0.f;
    for (int f=0;f<FF;++f){ h0+=fmean[f]*sw1[f*2+0]; h1+=fmean[f]*sw1[f*2+1]; }
    h0=fmaxf(h0,0.f); h1=fmaxf(h1,0.f);
    for (int f=0;f<FF;++f) g5[f]=sigm(h0*sw2[f]+h1*sw2[FF+f]);
  }

  // final adjacency: elu(sum_f elu(recon*(1+g))) row-normalized (overwrites p staging)
  {
    float nrmA=0.f;
    for (int m=0;m<NNODE;++m){
      float acc=0.f;
      for (int f=0;f<FF;++f) acc += eluf(s_adj[f*NNODE*NNODE+lane*NNODE+m]*(1.f+g5[f]));
      float e = eluf(acc);
      s_A[lane*NNODE+m]=e; nrmA+=fabsf(e);
    }
    float invA=1.f/(nrmA+1e-6f);
    for (int m=0;m<NNODE;++m) s_A[lane*NNODE+m]*=invA;
  }
  __syncthreads();

  // ---- ResGCN layer 1: x1 = selu(A @ (x@w1)) via WMMA ----
  {
    const float* w1 = rg_w1 + k*FF*15;
    for (int j=0;j<16;++j){
      float a=0.f;
      if (j<15) for (int f=0;f<FF;++f) a+=xr5[f]*w1[f*15+j];
      s_xw[lane*NNODE+j]=a;
    }
  }
  for (int f=0;f<FF;++f) s_cat[lane*52+f]=xr5[f];     // concat part 1: x
  __syncthreads();
  for (int mt=0;mt<2;++mt){
    v8f_t acc = zero8();
    for (int kk=0;kk<NNODE;kk+=4){
      v2f_t a = lda_frag(s_A,  NNODE, mt*16, kk, lane);
      v2f_t bb= ldb_frag(s_xw, NNODE, kk,    0,  lane);
      acc = wmma4(a,bb,acc);
    }
    int half=lane>>4, col=lane&15;
    #pragma unroll
    for (int i=0;i<8;++i){
      int row=mt*16+i+8*half;
      if (col<15) s_cat[row*52+5+col]=seluf(acc[i]);
    }
  }
  __syncthreads();
  cbam_c32(s_cat,52,20, cb1_w1+k*NNODE*8, cb1_w2+k*8*NNODE, cb1_cv+k*14, s_scr, lane);

  // ---- ResGCN layer 2: x2 = selu(A @ (x1@w2)) via WMMA ----
  {
    const float* w2 = rg_w2 + k*20*30;
    for (int j=0;j<NNODE;++j){
      float a=0.f;
      if (j<30) for (int c=0;c<20;++c) a+=s_cat[lane*52+c]*w2[c*30+j];
      s_xw[lane*NNODE+j]=a;
    }
  }
  __syncthreads();
  for (int nt=0;nt<2;++nt){
    for (int mt=0;mt<2;++mt){
      v8f_t acc = zero8();
      for (int kk=0;kk<NNODE;kk+=4){
        v2f_t a = lda_frag(s_A,  NNODE, mt*16, kk,    lane);
        v2f_t bb= ldb_frag(s_xw, NNODE, kk,    nt*16, lane);
        acc = wmma4(a,bb,acc);
      }
      int half=lane>>4, col=lane&15;
      #pragma unroll
      for (int i=0;i<8;++i){
        int row=mt*16+i+8*half, cg=nt*16+col;
        if (cg<30) s_cat[row*52+20+cg]=seluf(acc[i]);
      }
    }
  }
  __syncthreads();
  cbam_c32(s_cat,52,50, cb2_w1+k*NNODE*8, cb2_w2+k*8*NNODE, cb2_cv+k*14, s_scr, lane);

  // xi = elu(...) -> xr[b][k][n][d]
  {
    float* dst = xr_out + ((size_t)(b*KC+k))*NNODE*XRD;
    for (int idx=lane; idx<NNODE*XRD; idx+=32){
      int n=idx/XRD, d=idx%XRD;
      dst[idx]=eluf(s_cat[n*52+d]);
    }
  }
}

// ================= K2: attention coarsening per (b,k) =================
__global__ __launch_bounds__(32)
void k2_att(const float* __restrict__ xr, const float* __restrict__ Watt,
            float* __restrict__ xc_out)
{
  const int lane=threadIdx.x;
  const int b=blockIdx.x, k=blockIdx.y;
  __shared__ float s_xr[NNODE*52];   // 32x50 padded K->52
  __shared__ float s_w[52*80];
  __shared__ float s_fw[NNODE*80];
  __shared__ float s_s[96];

  const float* src = xr + ((size_t)(b*KC+k))*NNODE*XRD;
#if USE_ASYNC_LDS
  // stage W_att and xr via async global->LDS (ASYNCcnt) on contiguous words
  for (int i=lane;i<50*80;i+=32) async_copy_b32(&s_w[i], Watt+i);
  for (int i=lane;i<2*80;i+=32)  s_w[50*80+i]=0.f;
  for (int i=lane;i<NNODE*XRD;i+=32)
    async_copy_b32(&s_xr[(i/XRD)*52 + (i%XRD)], src+i);
  s_xr[lane*52+50]=0.f; s_xr[lane*52+51]=0.f;
  async_wait0();
#else
  for (int i=lane;i<NNODE*XRD;i+=32) s_xr[(i/XRD)*52 + (i%XRD)] = src[i];
  s_xr[lane*52+50]=0.f; s_xr[lane*52+51]=0.f;
  for (int i=lane;i<50*80;i+=32) s_w[i]=Watt[i];
  for (int i=lane;i<2*80;i+=32)  s_w[50*80+i]=0.f;
#endif
  __syncthreads();

  // fw = xr @ W_att  (32x52 @ 52x80, WMMA f32 16x16x4)
  for (int nt=0;nt<5;++nt){
    for (int mt=0;mt<2;++mt){
      v8f_t acc = zero8();
      for (int kk=0;kk<52;kk+=4){
        v2f_t a = lda_frag(s_xr, 52, mt*16, kk,    lane);
        v2f_t bb= ldb_frag(s_w,  80, kk,    nt*16, lane);
        acc = wmma4(a,bb,acc);
      }
      int half=lane>>4,col=lane&15;
      #pragma unroll
      for (int i=0;i<8;++i) s_fw[(mt*16+i+8*half)*80 + nt*16+col]=acc[i];
    }
  }
  __syncthreads();
  // att[n] = fw[n] . colsum(fw)
  for (int h=lane;h<80;h+=32){
    float a=0.f; for (int n=0;n<NNODE;++n) a+=s_fw[n*80+h];
    s_s[h]=a;
  }
  __syncthreads();
  float att=0.f;
  for (int h=0;h<80;++h) att+=s_fw[lane*80+h]*s_s[h];
  float mx=wave_max(att);
  float e=expf(att-mx);
  float den=wave_sum(e);
  float p=e/den;
  __syncthreads();
  s_s[lane]=p;
  __syncthreads();
  float* dst = xc_out + ((size_t)(b*KC+k))*XRD;
  for (int d=lane; d<XRD; d+=32){
    float a=0.f; for (int n=0;n<NNODE;++n) a+=s_s[n]*s_xr[n*52+d];
    dst[d]=a;
  }
}

// ================= K3: coarse stage per batch sample =================
__global__ __launch_bounds__(32)
void k3_coarse(const float* __restrict__ xc,
               const float* __restrict__ co_p,     const float* __restrict__ co_bias,
               const float* __restrict__ co_q,     const float* __restrict__ co_theta,
               const float* __restrict__ co_se_w1, const float* __restrict__ co_se_w2,
               const float* __restrict__ co_cb,    const float* __restrict__ co_win,
               const float* __restrict__ co_wout,
               const float* __restrict__ co_rg_w1, const float* __restrict__ co_rg_w2,
               const float* __restrict__ co_cb1_w1,const float* __restrict__ co_cb1_w2, const float* __restrict__ co_cb1_cv,
               const float* __restrict__ co_cb2_w1,const float* __restrict__ co_cb2_w2, const float* __restrict__ co_cb2_cv,
               float* __restrict__ out, float* __restrict__ l2_out)
{
  const int lane=threadIdx.x;
  const int b=blockIdx.x;
  __shared__ float s_xc[KC*COF];
  __shared__ float s_o[KC*COF];
  __shared__ float s_t[KC*COF];
  __shared__ float s_adjc[COF*KC*KC];   // adj channels -> later recon
  __shared__ float s_cbT[COE*COK];      // codebook transposed [32][128]
  __shared__ float s_cbsq[COK];
  __shared__ float s_cwin[KC*COE];
  __shared__ float s_cwout[COE*KC];
  __shared__ float s_ze16[16*COE];      // 16-row ze chunk
  __shared__ float s_zez16[16];
  __shared__ int   s_idx16[16];
  __shared__ float s_fsum[COF];
  __shared__ float s_h[28];
  __shared__ float s_g50[COF];
  __shared__ float s_ac[KC*KC];
  __shared__ float s_adjco[KC*KC];
  __shared__ float s_xw[KC*ODIM];
  __shared__ float s_x1[KC*ODIM];
  __shared__ float s_x1o[KC*ODIM];
  __shared__ float s_scr[216];

  __builtin_prefetch(co_theta, 0, 1);
  __builtin_prefetch(co_cb, 0, 1);
  const float* xcb = xc + (size_t)b*KC*COF;
  for (int i=lane;i<KC*COF;i+=32) s_xc[i]=xcb[i];
  for (int i=lane;i<COE*COK;i+=32){ int e=i/COK, n=i%COK; s_cbT[i]=co_cb[n*COE+e]; }
  for (int n=lane;n<COK;n+=32){
    float a=0.f; for (int e=0;e<COE;++e){ float v=co_cb[n*COE+e]; a+=v*v; }
    s_cbsq[n]=a;
  }
  for (int i=lane;i<KC*COE;i+=32) s_cwin[i]=co_win[i];
  for (int i=lane;i<COE*KC;i+=32) s_cwout[i]=co_wout[i];
  __syncthreads();

  // o = co_p @ xc + bias
  for (int idx=lane; idx<KC*COF; idx+=32){
    int i=idx/COF, d=idx%COF;
    float a=co_bias[idx];
    for (int j=0;j<KC;++j) a+=co_p[i*KC+j]*s_xc[j*COF+d];
    s_o[idx]=a;
  }
  __syncthreads();
  // t = o @ co_q
  for (int idx=lane; idx<KC*COF; idx+=32){
    int i=idx/COF, c=idx%COF;
    float a=0.f; for (int d=0;d<COF;++d) a+=s_o[i*COF+d]*co_q[d*COF+c];
    s_t[idx]=a;
  }
  __syncthreads();
  // g = elu(t @ theta) -> adjc[f][i][j]
  for (int idx=lane; idx<KC*KC*COF; idx+=32){
    int i=idx/(KC*COF), jj=idx%(KC*COF);
    float a=0.f;
    for (int c=0;c<COF;++c) a+=s_t[i*COF+c]*co_theta[c*(KC*COF)+jj];
    a=eluf(a);
    int j=jj/COF, f=jj%COF;
    s_adjc[f*(KC*KC)+i*KC+j]=a;
  }
  __syncthreads();
  // L1-normalize 350 rows over j
  for (int r=lane;r<COF*KC;r+=32){
    float* row=&s_adjc[r*KC];
    float nr=0.f; for (int j=0;j<KC;++j) nr+=fabsf(row[j]);
    float inv=1.f/(nr+1e-6f);
    for (int j=0;j<KC;++j) row[j]*=inv;
  }
  __syncthreads();

  // ---- VQ: ze = adj @ win (per 16-row chunk), WMMA distance GEMM vs 128 codes ----
  float sse=0.f;
  for (int mt=0; mt<22; ++mt){
    int r = mt*16 + (lane>>1);
    int e0 = (lane&1)*16;
    float zz=0.f;
    if (r < COF*KC){
      const float* arow=&s_adjc[r*KC];
      for (int e=e0;e<e0+16;++e){
        float a=0.f;
        for (int j=0;j<KC;++j) a+=arow[j]*s_cwin[j*COE+e];
        s_ze16[(lane>>1)*COE+e]=a; zz+=a*a;
      }
    } else {
      for (int e=e0;e<e0+16;++e) s_ze16[(lane>>1)*COE+e]=0.f;
    }
    zz += __shfl_xor(zz,1,32);
    if ((lane&1)==0) s_zez16[lane>>1]=zz;
    __syncthreads();

    float bd[8]; int bj[8];
    #pragma unroll
    for (int i=0;i<8;++i){ bd[i]=3.4e38f; bj[i]=0; }
    int half=lane>>4, col=lane&15;
    for (int nt=0;nt<8;++nt){
      v8f_t acc = zero8();
      for (int kk=0;kk<COE;kk+=4){
        v2f_t a = lda_frag(s_ze16, COE, 0,  kk,    lane);
        v2f_t bb= ldb_frag(s_cbT,  COK, kk, nt*16, lane);
        acc = wmma4(a,bb,acc);
      }
      int n=nt*16+col;
      #pragma unroll
      for (int i=0;i<8;++i){
        float d2 = s_zez16[i+8*half] - 2.f*acc[i] + s_cbsq[n];
        if (d2<bd[i] || (d2==bd[i] && n<bj[i])){ bd[i]=d2; bj[i]=n; }
      }
    }
    #pragma unroll
    for (int off=1; off<16; off<<=1){
      #pragma unroll
      for (int i=0;i<8;++i){
        float od=__shfl_xor(bd[i],off,32);
        int   oj=__shfl_xor(bj[i],off,32);
        if (od<bd[i] || (od==bd[i]&&oj<bj[i])){ bd[i]=od; bj[i]=oj; }
      }
    }
    if (col==0){
      #pragma unroll
      for (int i=0;i<8;++i) s_idx16[i+8*half]=bj[i];
    }
    __syncthreads();
    if (lane<16){
      int rr=mt*16+lane;
      if (rr<COF*KC){
        int j=s_idx16[lane];
        float s=0.f;
        for (int e=0;e<COE;++e){ float d=s_ze16[lane*COE+e]-s_cbT[e*COK+j]; s+=d*d; }
        sse+=s;
        for (int jc=0;jc<KC;++jc){
          float rsum=0.f;
          for (int e=0;e<COE;++e) rsum+=s_cbT[e*COK+j]*s_cwout[e*KC+jc];
          s_adjc[rr*KC+jc]=rsum;   // overwrite with recon
        }
      }
    }
    __syncthreads();
  }
  {
    float tot=wave_sum(sse);
    if (lane==0) l2_out[b]=tot;
  }

  // SE over 50 channels of recon
  for (int f=lane; f<COF; f+=32){
    float a=0.f; for (int t=0;t<KC*KC;++t) a+=s_adjc[f*(KC*KC)+t];
    s_fsum[f]=a*(1.f/(KC*KC));
  }
  __syncthreads();
  if (lane<25){
    float a=0.f; for (int f=0;f<COF;++f) a+=s_fsum[f]*co_se_w1[f*25+lane];
    s_h[lane]=fmaxf(a,0.f);
  }
  __syncthreads();
  for (int f=lane; f<COF; f+=32){
    float a=0.f; for (int p=0;p<25;++p) a+=s_h[p]*co_se_w2[p*COF+f];
    s_g50[f]=sigm(a);
  }
  __syncthreads();
  for (int idx=lane; idx<KC*KC; idx+=32){
    float a=0.f;
    for (int f=0;f<COF;++f) a+=eluf(s_adjc[f*(KC*KC)+idx]*(1.f+s_g50[f]));
    s_ac[idx]=eluf(a);
  }
  __syncthreads();
  if (lane<KC){
    float nr=0.f; for (int j=0;j<KC;++j) nr+=fabsf(s_ac[lane*KC+j]);
    float inv=1.f/(nr+1e-6f);
    for (int j=0;j<KC;++j) s_adjco[lane*KC+j]=s_ac[lane*KC+j]*inv;
  }
  __syncthreads();

  // ---- res_gcn2 ----
  for (int idx=lane; idx<KC*55; idx+=32){
    int i=idx/55, d=idx%55;
    float a=0.f; for (int c=0;c<COF;++c) a+=s_xc[i*COF+c]*co_rg_w1[c*55+d];
    s_xw[i*ODIM+d]=a;
  }
  __syncthreads();
  for (int idx=lane; idx<KC*55; idx+=32){
    int i=idx/55, d=idx%55;
    float a=0.f; for (int j=0;j<KC;++j) a+=s_adjco[i*KC+j]*s_xw[j*ODIM+d];
    a=seluf(a);
    s_x1[i*ODIM+d]=a; s_x1o[i*ODIM+d]=a;
  }
  __syncthreads();
  cbam_c7(s_x1,ODIM,55, co_cb1_w1, co_cb1_w2, co_cb1_cv, s_scr, lane);
  for (int idx=lane; idx<KC*55; idx+=32){ int i=idx/55,d=idx%55; s_x1[i*ODIM+d]+=s_x1o[i*ODIM+d]; }
  __syncthreads();

  for (int idx=lane; idx<KC*ODIM; idx+=32){
    int i=idx/ODIM, d=idx%ODIM;
    float a=0.f; for (int c=0;c<55;++c) a+=s_x1[i*ODIM+c]*co_rg_w2[c*ODIM+d];
    s_xw[i*ODIM+d]=a;
  }
  __syncthreads();
  for (int idx=lane; idx<KC*ODIM; idx+=32){
    int i=idx/ODIM, d=idx%ODIM;
    float a=0.f; for (int j=0;j<KC;++j) a+=s_adjco[i*KC+j]*s_xw[j*ODIM+d];
    a=seluf(a);
    s_x1[i*ODIM+d]=a; s_x1o[i*ODIM+d]=a;
  }
  __syncthreads();
  cbam_c7(s_x1,ODIM,ODIM, co_cb2_w1, co_cb2_w2, co_cb2_cv, s_scr, lane);

  float* dst = out + 1 + (size_t)b*KC*ODIM;
  for (int idx=lane; idx<KC*ODIM; idx+=32){
    int i=idx/ODIM,d=idx%ODIM;
    dst[idx]=s_x1[i*ODIM+d]+s_x1o[i*ODIM+d];
  }
}

// ================= K4: deterministic loss reduction =================
__global__ __launch_bounds__(256)
void k4_loss(const float* __restrict__ l1, const float* __restrict__ l2,
             float* __restrict__ out)
{
  __shared__ float red[256];
  int t=threadIdx.x;
  float a=0.f;
  for (int i=t;i<KC*BATCH;i+=256) a+=l1[i];
  float bsum=0.f;
  for (int i=t;i<BATCH;i+=256) bsum+=l2[i];
  // losses.sum(): per cluster 1.25*SSE/(B*5*32*16); + 2 * 1.25*SSE_co/(B*50*7*32)
  const float c1 = 1.25f/((float)BATCH*FF*NNODE*EE);
  const float c2 = 2.f*1.25f/((float)BATCH*COF*KC*COE);
  red[t]=a*c1 + bsum*c2;
  __syncthreads();
  for (int s=128;s>0;s>>=1){ if (t<s) red[t]+=red[t+s]; __syncthreads(); }
  if (t==0) out[0]=red[0];
}

// ================= host launcher =================
extern "C" void kernel_launch(void* const* d_in, const int* in_sizes, int n_in,
                              void* d_out, int out_size, void* d_ws, size_t ws_size,
                              hipStream_t stream)
{
  (void)in_sizes; (void)n_in; (void)out_size; (void)ws_size;
  const float* x        = (const float*)d_in[0];
  const float* ga_p     = (const float*)d_in[1];
  const float* ga_bias  = (const float*)d_in[2];
  const float* ga_q     = (const float*)d_in[3];
  const float* ga_theta = (const float*)d_in[4];
  const float* ga_se_w1 = (const float*)d_in[5];
  const float* ga_se_w2 = (const float*)d_in[6];
  const float* ga_cb    = (const float*)d_in[7];
  const float* ga_win   = (const float*)d_in[8];
  const float* ga_wout  = (const float*)d_in[9];
  const float* rg_w1    = (const float*)d_in[10];
  const float* rg_w2    = (const float*)d_in[11];
  const float* cb1_w1   = (const float*)d_in[12];
  const float* cb1_w2   = (const float*)d_in[13];
  const float* cb1_cv   = (const float*)d_in[14];
  const float* cb2_w1   = (const float*)d_in[15];
  const float* cb2_w2   = (const float*)d_in[16];
  const float* cb2_cv   = (const float*)d_in[17];
  const float* co_p     = (const float*)d_in[18];
  const float* co_bias  = (const float*)d_in[19];
  const float* co_q     = (const float*)d_in[20];
  const float* co_theta = (const float*)d_in[21];
  const float* co_se_w1 = (const float*)d_in[22];
  const float* co_se_w2 = (const float*)d_in[23];
  const float* co_cb    = (const float*)d_in[24];
  const float* co_win   = (const float*)d_in[25];
  const float* co_wout  = (const float*)d_in[26];
  const float* co_rg_w1 = (const float*)d_in[27];
  const float* co_rg_w2 = (const float*)d_in[28];
  const float* co_cb1_w1= (const float*)d_in[29];
  const float* co_cb1_w2= (const float*)d_in[30];
  const float* co_cb1_cv= (const float*)d_in[31];
  const float* co_cb2_w1= (const float*)d_in[32];
  const float* co_cb2_w2= (const float*)d_in[33];
  const float* co_cb2_cv= (const float*)d_in[34];
  const float* W_att    = (const float*)d_in[35];

  // workspace layout (floats): xr | xc | l1 | l2  (~47.4 MB total)
  float* ws = (float*)d_ws;
  float* xr = ws;
  float* xc = xr + (size_t)BATCH*KC*NNODE*XRD;
  float* l1 = xc + (size_t)BATCH*KC*XRD;
  float* l2 = l1 + (size_t)KC*BATCH;
  float* out = (float*)d_out;

  dim3 gbk(BATCH, KC);
  k1_cluster<<<gbk, 32, 0, stream>>>(x, ga_p, ga_bias, ga_q, ga_theta,
                                     ga_se_w1, ga_se_w2, ga_cb, ga_win, ga_wout,
                                     rg_w1, rg_w2, cb1_w1, cb1_w2, cb1_cv,
                                     cb2_w1, cb2_w2, cb2_cv, xr, l1);
  k2_att<<<gbk, 32, 0, stream>>>(xr, W_att, xc);
  k3_coarse<<<BATCH, 32, 0, stream>>>(xc, co_p, co_bias, co_q, co_theta,
                                      co_se_w1, co_se_w2, co_cb, co_win, co_wout,
                                      co_rg_w1, co_rg_w2,
                                      co_cb1_w1, co_cb1_w2, co_cb1_cv,
                                      co_cb2_w1, co_cb2_w2, co_cb2_cv,
                                      out, l2);
  k4_loss<<<1, 256, 0, stream>>>(l1, l2, out);
}